// Attention_18769007083684
// MI455X (gfx1250) — compile-verified
//
#include <hip/hip_runtime.h>
#include <hip/hip_bf16.h>
#include <math.h>

typedef __attribute__((ext_vector_type(2))) float v2f;
typedef __attribute__((ext_vector_type(8))) float v8f;

#define BB 2
#define TT 2048
#define DD 1024
#define HH 16
#define DHH 64
#define NEG_INF (-INFINITY)

static __device__ __forceinline__ v8f wmma4(v2f a, v2f b, v8f c) {
    // V_WMMA_F32_16X16X4_F32 : D = A(16x4) * B(4x16) + C(16x16)
    return __builtin_amdgcn_wmma_f32_16x16x4_f32(
        false, a, false, b, (short)0, c, false, false);
}

static __device__ __forceinline__ float redmax16(float v) {
    #pragma unroll
    for (int off = 1; off < 16; off <<= 1) v = fmaxf(v, __shfl_xor(v, off, 32));
    return v;
}
static __device__ __forceinline__ float redsum16(float v) {
    #pragma unroll
    for (int off = 1; off < 16; off <<= 1) v += __shfl_xor(v, off, 32);
    return v;
}

static __device__ __forceinline__ void async_load_b128_to_lds(void* lds_dst,
                                                              const void* gsrc) {
    unsigned lds_off = (unsigned)(size_t)lds_dst;     // addrspace(3) byte offset
    unsigned long long ga = (unsigned long long)gsrc;
    asm volatile("global_load_async_to_lds_b128 %0, %1, off"
                 :: "v"(lds_off), "v"(ga) : "memory");
}

// ---------------------------------------------------------------------------
// NT GEMM: C[M,N] = A[M,K] * B[N,K]^T  (row-major, fp32 WMMA).
// Block = 8 waves computing a 128(M) x 64(N) tile; the shared 64-row B strip
// is staged through LDS with double-buffered async copies (ASYNCcnt).
// Each wave owns a 16x64 strip: 4 accumulators, A k-chunk preloaded to regs.
// ---------------------------------------------------------------------------
#define KC 32   // k-chunk (floats)

__global__ __launch_bounds__(256) void gemm_nt_f32_wmma(
    const float* __restrict__ A, const float* __restrict__ Bm,
    float* __restrict__ C, int M, int N, int K, int lda, int ldb, int ldc)
{
    __shared__ float bsh[2][64][KC];

    const int tid  = (int)threadIdx.x;
    const int w    = tid >> 5;
    const int lane = tid & 31;
    const int nblk = N >> 6;
    const int mb   = (int)blockIdx.x / nblk;     // 128-row block index
    const int nb   = (int)blockIdx.x % nblk;
    const int m0   = mb * 128 + w * 16;
    const int n0   = nb * 64;

    const int r16 = lane & 15;
    const int hi2 = (lane < 16) ? 0 : 2;
    const int hi8 = (lane < 16) ? 0 : 8;

    // async-copy assignment: 2 x b128 per thread per chunk
    const int idx0 = tid;            // covers rows 0..31 (8 segs/row)
    const int idx1 = tid + 256;      // covers rows 32..63
    const int brow0 = idx0 >> 3, bseg0 = idx0 & 7;
    const int brow1 = idx1 >> 3, bseg1 = idx1 & 7;
    const float* gB0 = Bm + (size_t)(n0 + brow0) * ldb + bseg0 * 4;
    const float* gB1 = Bm + (size_t)(n0 + brow1) * ldb + bseg1 * 4;

    v8f acc[4];
    #pragma unroll
    for (int n = 0; n < 4; ++n) acc[n] = (v8f){0.f,0.f,0.f,0.f,0.f,0.f,0.f,0.f};

    const float* arow = A + (size_t)(m0 + r16) * lda + hi2;
    const int NC = K / KC;

    // prologue: stage chunk 0
    async_load_b128_to_lds(&bsh[0][brow0][bseg0 * 4], gB0);
    async_load_b128_to_lds(&bsh[0][brow1][bseg1 * 4], gB1);

    for (int c = 0; c < NC; ++c) {
        const int kc  = c * KC;
        const int buf = c & 1;
        if (c + 1 < NC) {
            async_load_b128_to_lds(&bsh[buf ^ 1][brow0][bseg0 * 4], gB0 + kc + KC);
            async_load_b128_to_lds(&bsh[buf ^ 1][brow1][bseg1 * 4], gB1 + kc + KC);
            asm volatile("s_wait_asynccnt 0x2" ::: "memory");
        } else {
            asm volatile("s_wait_asynccnt 0x0" ::: "memory");
        }
        __syncthreads();

        // preload this wave's A chunk (8 k-steps) into registers
        v2f aq[KC / 4];
        __builtin_prefetch(arow + kc + 2 * KC, 0, 3);
        #pragma unroll
        for (int j = 0; j < KC / 4; ++j) aq[j] = *(const v2f*)(arow + kc + j * 4);

        #pragma unroll
        for (int j = 0; j < KC / 4; ++j) {
            #pragma unroll
            for (int n = 0; n < 4; ++n) {
                v2f bv = *(const v2f*)&bsh[buf][n * 16 + r16][j * 4 + hi2];
                acc[n] = wmma4(aq[j], bv, acc[n]);
            }
        }
        __syncthreads();   // protect buf before it is overwritten next issue
    }

    #pragma unroll
    for (int n = 0; n < 4; ++n)
        #pragma unroll
        for (int r = 0; r < 8; ++r)
            C[(size_t)(m0 + r + hi8) * ldc + n0 + n * 16 + r16] = acc[n][r];
}

// ---------------------------------------------------------------------------
// Fused causal attention with prior. One wave per (b, h, 16-row q tile).
// ---------------------------------------------------------------------------
__global__ __launch_bounds__(128) void attn_fused_wmma(
    const float* __restrict__ qkv,           // (B,T,3,H,dh)
    const float* __restrict__ prior,         // (B,T,T)
    const unsigned char* __restrict__ qmask, // (B,T)
    float* __restrict__ prob,                // (B,H,T,T) output (+ Z scratch)
    float* __restrict__ score,               // (B,H,T,T) output
    float* __restrict__ yat)                 // (B,T,D) workspace
{
    __shared__ float sm[4][16];
    __shared__ float si[4][16];

    const int w    = (int)(threadIdx.x >> 5);
    const int lane = (int)(threadIdx.x & 31);
    const int wid  = (int)blockIdx.x * 4 + w;

    const int b  = wid >> 11;
    const int h  = (wid >> 7) & 15;
    const int qt = wid & 127;
    const int q0 = qt * 16;

    const int r16 = lane & 15;
    const int hi2 = (lane < 16) ? 0 : 2;
    const int hi8 = (lane < 16) ? 0 : 8;

    const size_t row_stride = 3 * DD;
    const float* qbase = qkv + (size_t)b * TT * row_stride + h * DHH;
    const float* kbase = qbase + DD;
    const float* vbase = qbase + 2 * DD;
    const float* prbase = prior + (size_t)b * TT * TT;
    float* sc_base = score + ((size_t)(b * HH + h) * TT) * TT;
    float* pr_base = prob  + ((size_t)(b * HH + h) * TT) * TT;

    bool qm[8];
    #pragma unroll
    for (int r = 0; r < 8; ++r) qm[r] = qmask[(size_t)b * TT + q0 + r + hi8] != 0;

    float m_run[8], s_run[8];
    #pragma unroll
    for (int r = 0; r < 8; ++r) { m_run[r] = NEG_INF; s_run[r] = 0.f; }

    // hoist the Q fragment: loop-invariant across all kv tiles (16 v2f regs)
    const float* qrow = qbase + (size_t)(q0 + r16) * row_stride + hi2;
    v2f aq[DHH / 4];
    #pragma unroll
    for (int kk = 0; kk < DHH / 4; ++kk) aq[kk] = *(const v2f*)(qrow + kk * 4);

    // ---------------- Pass 1: scores, Z, online stats ----------------
    for (int kt = 0; kt < TT / 16; ++kt) {
        const int kv0 = kt * 16;
        const int col = kv0 + r16;
        if (kt > qt) {   // fully causal-masked tile: constants only
            #pragma unroll
            for (int r = 0; r < 8; ++r) {
                const int row = q0 + r + hi8;
                __builtin_nontemporal_store(NEG_INF, &sc_base[(size_t)row * TT + col]);
                __builtin_nontemporal_store(0.f,     &pr_base[(size_t)row * TT + col]);
            }
            continue;
        }
        v8f c = (v8f){0.f,0.f,0.f,0.f,0.f,0.f,0.f,0.f};
        const float* krow = kbase + (size_t)(kv0 + r16) * row_stride + hi2;
        #pragma unroll
        for (int kk = 0; kk < DHH / 4; ++kk) {
            v2f bb = *(const v2f*)(krow + kk * 4);
            c = wmma4(aq[kk], bb, c);
        }
        #pragma unroll
        for (int r = 0; r < 8; ++r) {
            const int row = q0 + r + hi8;
            const float sc = c[r] * 0.125f;              // * dh^-0.5
            const bool masked = qm[r] || (col > row);
            __builtin_nontemporal_store(masked ? NEG_INF : sc,
                                        &sc_base[(size_t)row * TT + col]);
            float z = masked ? NEG_INF
                             : sc + __logf(prbase[(size_t)row * TT + col] + 1e-8f);
            pr_base[(size_t)row * TT + col] = z;         // Z scratch (re-read)
            const float tmax = redmax16(z);
            const float mnew = fmaxf(m_run[r], tmax);
            const float pe   = (z > NEG_INF && mnew > NEG_INF) ? __expf(z - mnew) : 0.f;
            const float tsum = redsum16(pe);
            const float sfac = (m_run[r] > NEG_INF) ? __expf(m_run[r] - mnew) : 0.f;
            s_run[r] = s_run[r] * sfac + tsum;
            m_run[r] = mnew;
        }
    }

    // Exchange per-row stats across lane halves via tiny LDS table.
    #pragma unroll
    for (int r = 0; r < 8; ++r) {
        const float is = (m_run[r] > NEG_INF && s_run[r] > 0.f) ? 1.0f / s_run[r] : 0.f;
        if (lane == r)      { sm[w][r]     = m_run[r]; si[w][r]     = is; }
        if (lane == 16 + r) { sm[w][8 + r] = m_run[r]; si[w][8 + r] = is; }
    }
    __syncthreads();
    const float m_row  = sm[w][r16];
    const float is_row = si[w][r16];

    // ---------------- Pass 2: normalize p, accumulate y = P * V ----------------
    v8f acc[4];
    #pragma unroll
    for (int n = 0; n < 4; ++n) acc[n] = (v8f){0.f,0.f,0.f,0.f,0.f,0.f,0.f,0.f};

    for (int kt = 0; kt <= qt; ++kt) {
        #pragma unroll
        for (int j = 0; j < 4; ++j) {
            const int cb = kt * 16 + 4 * j + hi2;
            float* zp = pr_base + (size_t)(q0 + r16) * TT + cb;
            v2f z2 = *(const v2f*)zp;
            v2f p2;
            p2.x = (z2.x > NEG_INF && m_row > NEG_INF) ? __expf(z2.x - m_row) * is_row : 0.f;
            p2.y = (z2.y > NEG_INF && m_row > NEG_INF) ? __expf(z2.y - m_row) * is_row : 0.f;
            __builtin_nontemporal_store(p2, (v2f*)zp);    // final attn_prob
            const float* vr = vbase + (size_t)cb * row_stride;
            #pragma unroll
            for (int n = 0; n < 4; ++n) {
                v2f bv;
                bv.x = vr[n * 16 + r16];
                bv.y = vr[row_stride + n * 16 + r16];
                acc[n] = wmma4(p2, bv, acc[n]);
            }
        }
    }
    #pragma unroll
    for (int n = 0; n < 4; ++n)
        #pragma unroll
        for (int r = 0; r < 8; ++r)
            yat[((size_t)b * TT + q0 + r + hi8) * DD + h * DHH + n * 16 + r16] = acc[n][r];
}

// ---------------------------------------------------------------------------
extern "C" void kernel_launch(void* const* d_in, const int* in_sizes, int n_in,
                              void* d_out, int out_size, void* d_ws, size_t ws_size,
                              hipStream_t stream) {
    (void)in_sizes; (void)n_in; (void)out_size; (void)ws_size;
    const float*         query = (const float*)d_in[0];
    const unsigned char* qmask = (const unsigned char*)d_in[1];
    const float*         prior = (const float*)d_in[2];
    const float*         Wqkv  = (const float*)d_in[3];
    const float*         Wo    = (const float*)d_in[4];

    float* y     = (float*)d_out;
    float* prob  = y + (size_t)BB * TT * DD;
    float* score = prob + (size_t)BB * HH * TT * TT;

    float* qkv = (float*)d_ws;                       // (B,T,3,H,dh)
    float* yat = qkv + (size_t)BB * TT * 3 * DD;     // (B,T,D)

    // K1: qkv = query @ Wqkv^T   (M=4096, N=3072, K=1024)
    {
        const int M = BB * TT, N = 3 * DD, K = DD;
        const int blocks = (M / 128) * (N / 64);
        gemm_nt_f32_wmma<<<blocks, 256, 0, stream>>>(query, Wqkv, qkv,
                                                     M, N, K, DD, DD, 3 * DD);
    }
    // K2: fused attention (4096 waves, 4 per block)
    {
        const int waves = BB * HH * (TT / 16);
        attn_fused_wmma<<<waves / 4, 128, 0, stream>>>(qkv, prior, qmask,
                                                       prob, score, yat);
    }
    // K3: y = yat @ Wo^T   (M=4096, N=1024, K=1024)
    {
        const int M = BB * TT, N = DD, K = DD;
        const int blocks = (M / 128) * (N / 64);
        gemm_nt_f32_wmma<<<blocks, 256, 0, stream>>>(yat, Wo, y,
                                                     M, N, K, DD, DD, DD);
    }
}